// TT_conv_49100066128381
// MI455X (gfx1250) — compile-verified
//
#include <hip/hip_runtime.h>
#include <hip/hip_bf16.h>

typedef __attribute__((ext_vector_type(2))) float v2f;
typedef __attribute__((ext_vector_type(8))) float v8f;

// D(16x16) = A(16x4) * B(4x16) + C, all f32, wave32.
// Per-lane layout (ISA 7.12.2): half = lane/16, idx = lane%16
//   A: a.x = A[M=idx][K=2*half+0], a.y = A[M=idx][K=2*half+1]
//   B: b.x = B[K=2*half+0][N=idx], b.y = B[K=2*half+1][N=idx]
//   C/D: c[r] = D[M=r+8*half][N=idx]
__device__ __forceinline__ v8f wmma4(v2f a, v2f b, v8f c) {
  return __builtin_amdgcn_wmma_f32_16x16x4_f32(false, a, false, b, (short)0, c,
                                               false, false);
}

#define XT_PITCH 68
// LDS offset for im2col K index (ci0,kh,kw) = (K/9, (K%9)/3, K%3).
// Called only with post-unroll constants -> fully constant-folded.
__device__ __forceinline__ constexpr int xoff(int K) {
  return (K / 9) * (6 * XT_PITCH) + ((K % 9) / 3) * XT_PITCH + ((K % 9) % 3);
}

// ---------------------------------------------------------------------------
// Pass 0: build TT matrix W[64][64] and pre-swizzle W + conv kernel into the
// per-lane WMMA B/A fragment layouts.
//   Wsw[((ks*4+n)*32+lane)*2+t] = W[4*ks+2*half+t][n*16+idx]   (B for pass 2)
//   Ksw[(ks*32+lane)*2+t]       = Ker[co0=idx][4*ks+2*half+t]  (A for pass 1)
// where Ker[co0][K], K = ci0*9 + kh*3 + kw.
// ---------------------------------------------------------------------------
__global__ __launch_bounds__(256) void tt_prep(
    const float* __restrict__ c0, const float* __restrict__ c1,
    const float* __restrict__ c2, const float* __restrict__ kflat,
    float* __restrict__ Wsw, float* __restrict__ Ksw) {
  for (int e = threadIdx.x; e < 4096; e += 256) {
    int i = e >> 6, j = e & 63;                       // i=cin row, j=co col
    int m1 = i >> 4, m2 = (i >> 2) & 3, m3 = i & 3;
    int n1 = j >> 4, n2 = (j >> 2) & 3, n3 = j & 3;
    float s = 0.f;
    for (int r1 = 0; r1 < 8; ++r1) {
      float a = c0[(m1 * 4 + n1) * 8 + r1];           // core0 (1,4,4,8)
      for (int r2 = 0; r2 < 8; ++r2) {
        s += a * c1[((r1 * 4 + m2) * 4 + n2) * 8 + r2]  // core1 (8,4,4,8)
               * c2[(r2 * 4 + m3) * 4 + n3];            // core2 (8,4,4,1)
      }
    }
    int ks = i >> 2, rem = i & 3, half = rem >> 1, t = rem & 1;
    int n = j >> 4, idx = j & 15, lane = half * 16 + idx;
    Wsw[((ks * 4 + n) * 32 + lane) * 2 + t] = s;
  }
  for (int e = threadIdx.x; e < 2304; e += 256) {     // 36 ksteps * 32 * 2
    int t = e & 1, lane = (e >> 1) & 31, ks = e >> 6;
    int idx = lane & 15, half = lane >> 4;
    int K = 4 * ks + 2 * half + t;                    // 0..143
    Ksw[e] = kflat[idx * 144 + K];                    // kernel (16,16,3,3)
  }
}

// ---------------------------------------------------------------------------
// Pass 1: conv as WMMA GEMM.  M=co0(16), N=position, K=(ci0,kh,kw)=144.
// Block = one (b,cin) image x one 4-row*64-col output tile (16 N-tiles).
// Y layout (cin-pair interleaved for b64 A-loads in pass 2):
//   Y[ ((b*32 + cin/2)*61504 + co0*3844 + i*62 + j)*2 + (cin&1) ]
// ---------------------------------------------------------------------------
__global__ __launch_bounds__(256) void conv_wmma(
    const float* __restrict__ x, const float* __restrict__ Ksw,
    float* __restrict__ Y) {
  __shared__ float xt[16 * 6 * XT_PITCH];   // 16 ci0 x 6 rows x 68 cols
  __shared__ float kl[2304];

  const int tid = threadIdx.x;
  const int blk = blockIdx.x;               // n*16 + rowblock
  const int rb  = blk & 15;
  const int n   = blk >> 4;                 // 0..511  (= b*64 + cin)
  const int b   = n >> 6, cin = n & 63;
  const float* xp = x + ((size_t)(b * 1024 + cin * 16)) * 4096; // [ci0][64][64]

  // stage x tile, coalesced by column; edge rows/cols clamped (clamped data
  // only feeds positions masked at store time)
  {
    const int c = tid & 63, q = tid >> 6;   // 64 cols x 4 row-groups
#pragma unroll
    for (int ci0 = 0; ci0 < 16; ++ci0) {
      for (int r = q; r < 6; r += 4) {
        int gr = rb * 4 + r; gr = gr > 63 ? 63 : gr;
        const float* row = xp + ci0 * 4096 + gr * 64;
        xt[ci0 * (6 * XT_PITCH) + r * XT_PITCH + c] = row[c];
        if (c < 2)  // columns 64,65 clamp to 63
          xt[ci0 * (6 * XT_PITCH) + r * XT_PITCH + 64 + c] = row[63];
      }
    }
  }
  for (int e = tid; e < 2304; e += 256) kl[e] = Ksw[e];
  __syncthreads();

  const int wave = tid >> 5, lane = tid & 31;
  const int half = lane >> 4, idx = lane & 15;
  const int p0 = (wave * 2) * 16 + idx;     // N-tile 2*wave
  const int p1 = p0 + 16;                   // N-tile 2*wave+1
  const int base0 = (p0 >> 6) * XT_PITCH + (p0 & 63); // di*pitch + dj
  const int base1 = (p1 >> 6) * XT_PITCH + (p1 & 63);

  v8f acc0 = {}, acc1 = {};
#pragma unroll
  for (int ks = 0; ks < 36; ++ks) {
    v2f a = *(const v2f*)&kl[(ks * 32 + lane) * 2];   // kernel A-frag
    // both half candidates are compile-time constants after unroll
    const int offx = half ? xoff(4 * ks + 2) : xoff(4 * ks + 0);
    const int offy = half ? xoff(4 * ks + 3) : xoff(4 * ks + 1);
    v2f b0, b1;
    b0.x = xt[offx + base0]; b0.y = xt[offy + base0];
    b1.x = xt[offx + base1]; b1.y = xt[offy + base1];
    acc0 = wmma4(a, b0, acc0);
    acc1 = wmma4(a, b1, acc1);
  }

  float* yp = Y + ((size_t)(b * 32 + (cin >> 1)) * 61504) * 2 + (cin & 1);
#pragma unroll
  for (int r = 0; r < 8; ++r) {
    const int co0 = r + 8 * half;
    {
      int di = p0 >> 6, dj = p0 & 63, i = rb * 4 + di, j = dj;
      if (i < 62 && j < 62) yp[(co0 * 3844 + i * 62 + j) * 2] = acc0[r];
    }
    {
      int di = p1 >> 6, dj = p1 & 63, i = rb * 4 + di, j = dj;
      if (i < 62 && j < 62) yp[(co0 * 3844 + i * 62 + j) * 2] = acc1[r];
    }
  }
}

// ---------------------------------------------------------------------------
// Pass 2: Y(492032 x 64) @ W(64 x 64) + bias -> d_out (exact output layout).
// Wave = one 16-row M-tile, all 4 N-tiles.  K = 64 cin -> 16 k-steps.
// A frag = one b64 load: Y[b][pair=2*ks+half][Rl][0..1], Rl = tl*16 + idx.
// ---------------------------------------------------------------------------
__global__ __launch_bounds__(256) void gemm_tt(
    const float* __restrict__ Y, const float* __restrict__ Wsw,
    const float* __restrict__ bias, float* __restrict__ out) {
  __shared__ float wl[4096];
  const int tid = threadIdx.x;
  for (int e = tid; e < 4096; e += 256) wl[e] = Wsw[e];
  __syncthreads();

  const int wave = tid >> 5, lane = tid & 31;
  const int half = lane >> 4, idx = lane & 15;
  const int T = blockIdx.x * 8 + wave;      // global M-tile, < 30752
  const int b = T / 3844;
  const int tl = T - b * 3844;
  const int Rl = tl * 16 + idx;             // row within batch slab
  const float* yb = Y + (size_t)b * 32 * 61504 * 2;

  v8f acc[4];
#pragma unroll
  for (int nn = 0; nn < 4; ++nn) {          // bias folded into C init
    float bv = bias[nn * 16 + idx];
    acc[nn] = (v8f){bv, bv, bv, bv, bv, bv, bv, bv};
  }

#pragma unroll
  for (int ks = 0; ks < 16; ++ks) {
    const int pair = 2 * ks + half;
    v2f a = *(const v2f*)&yb[((size_t)pair * 61504 + Rl) * 2];
#pragma unroll
    for (int nn = 0; nn < 4; ++nn) {
      v2f w = *(const v2f*)&wl[((ks * 4 + nn) * 32 + lane) * 2];
      acc[nn] = wmma4(a, w, acc[nn]);
    }
  }

  float* op = out + (size_t)T * 16 * 64;
#pragma unroll
  for (int r = 0; r < 8; ++r) {
    const int m = r + 8 * half;
#pragma unroll
    for (int nn = 0; nn < 4; ++nn) {
      op[m * 64 + nn * 16 + idx] = acc[nn][r];
    }
  }
}

extern "C" void kernel_launch(void* const* d_in, const int* in_sizes, int n_in,
                              void* d_out, int out_size, void* d_ws,
                              size_t ws_size, hipStream_t stream) {
  (void)in_sizes; (void)n_in; (void)out_size; (void)ws_size;
  const float* x      = (const float*)d_in[0]; // (8,1024,64,64)
  const float* kernel = (const float*)d_in[1]; // (16,16,3,3)
  const float* core0  = (const float*)d_in[2]; // (1,4,4,8)
  const float* core1  = (const float*)d_in[3]; // (8,4,4,8)
  const float* core2  = (const float*)d_in[4]; // (8,4,4,1)
  const float* bias   = (const float*)d_in[5]; // (64,)

  float* Wsw = (float*)d_ws;                          // 4096 floats
  float* Ksw = Wsw + 4096;                            // 2304 floats
  float* Yb  = (float*)((char*)d_ws + 32768);         // 31,490,048 floats

  tt_prep<<<1, 256, 0, stream>>>(core0, core1, core2, kernel, Wsw, Ksw);
  conv_wmma<<<512 * 16, 256, 0, stream>>>(x, Ksw, Yb);
  gemm_tt<<<3844, 256, 0, stream>>>(Yb, Wsw, bias, (float*)d_out);
}